// RectAverage_45251775431276
// MI455X (gfx1250) — compile-verified
//
#include <hip/hip_runtime.h>

typedef float v2f __attribute__((ext_vector_type(2)));
typedef float v8f __attribute__((ext_vector_type(8)));

#define IMG   512
#define HW    (IMG * IMG)          // 262144 pixels per channel
#define BATCH 16
#define NBIN  257                  // IMAGE_SIZE/2 + 1
#define BLOCKS_PER_BATCH 32
#define ROWS_PER_BLOCK   16        // 8 wave32s * 2 rows each
#define CHUNKS_PER_GROUP 8         // 8 x 16 pixels loaded ahead of the WMMAs

// ---------------------------------------------------------------------------
// Kernel 1: stream x once, luma via V_WMMA_F32_16X16X4_F32, analytic ring-bin
// segmented sum into an LDS histogram, deterministic per-block partial store.
// Loads for 8 chunks are issued as a batch so each wave keeps ~16 global
// loads in flight instead of draining loadcnt before every WMMA.
// ---------------------------------------------------------------------------
__global__ __launch_bounds__(256)
void rect_profile_kernel(const float* __restrict__ x, float* __restrict__ partial)
{
    __shared__ float bins[NBIN];

    const int tid  = threadIdx.x;
    const int b    = blockIdx.x / BLOCKS_PER_BATCH;   // batch
    const int blk  = blockIdx.x % BLOCKS_PER_BATCH;   // row-group within batch
    const int wave = tid >> 5;                        // wave32 id (0..7)
    const int lane = tid & 31;
    const int li   = lane & 15;                       // pixel slot within chunk (N)
    const int hs   = lane >> 4;                       // half-wave select (K pairs)

    for (int i = tid; i < NBIN; i += 256) bins[i] = 0.0f;
    __syncthreads();

    // A fragment (16x4 f32): all 16 M-rows hold the same scaled luma weights.
    // Layout: lanes 0-15 supply K=0,1 ; lanes 16-31 supply K=2,3.
    v2f a;
    a.x = hs ? (20.0f * 0.114f) : (20.0f * 0.299f);   // K2 : K0
    a.y = hs ? 0.0f             : (20.0f * 0.587f);   // K3 : K1

    const float* xb = x + (size_t)b * (3 * HW);
    // Channel offsets per half-wave. Upper half re-reads channel 2 for the K=3
    // slot; its A-weight is 0 so it cancels, and the address is always valid.
    const int c0 = hs ? 2 * HW : 0;
    const int c1 = hs ? 2 * HW : HW;

    for (int r = 0; r < 2; ++r) {
        const int h = blk * ROWS_PER_BLOCK + wave * 2 + r;
        const float* row = xb + (size_t)h * IMG;

        for (int g = 0; g < IMG; g += 16 * CHUNKS_PER_GROUP) {
            // ---- batch-issue all loads for this group (16 loads in flight)
            v2f buf[CHUNKS_PER_GROUP];
#pragma unroll
            for (int j = 0; j < CHUNKS_PER_GROUP; ++j) {
                const int w = g + 16 * j + li;
                buf[j].x = row[c0 + w];
                buf[j].y = row[c1 + w];
            }
            // ---- consume: WMMA luma + analytic ring-bin LDS accumulate
#pragma unroll
            for (int j = 0; j < CHUNKS_PER_GROUP; ++j) {
                const int w = g + 16 * j + li;
                v8f c = {};
                // D = A x B + 0 : every lane's c[0] = luma(h, g + 16j + li)
                c = __builtin_amdgcn_wmma_f32_16x16x4_f32(
                        /*neg_a=*/false, a, /*neg_b=*/false, buf[j],
                        /*c_mod=*/(short)0, c, /*reuse_a=*/false, /*reuse_b=*/false);

                // Analytic one-hot ring index (replaces the 269 MB mask read).
                int bin;
                if (h == 0 || w == 0) {
                    bin = NBIN - 1;                    // plane border = 256
                } else {
                    const int J = h - 1, K = w - 1;
                    const int ring = min(min(J, K), min(510 - J, 510 - K));
                    bin = 255 - ring;
                }
                if (hs == 0) atomicAdd(&bins[bin], c[0]);  // ds_add_f32
            }
        }
    }
    __syncthreads();

    float* dst = partial + ((size_t)b * BLOCKS_PER_BATCH + blk) * NBIN;
    for (int i = tid; i < NBIN; i += 256) dst[i] = bins[i];
}

// ---------------------------------------------------------------------------
// Kernel 2: deterministic partial-sum, divide by mask_n, global min/max,
// normalize. 16*257 = 4112 outputs -> single 512-thread block.
// ---------------------------------------------------------------------------
__global__ __launch_bounds__(512)
void rect_finalize_kernel(const float* __restrict__ partial,
                          const float* __restrict__ mask_n,
                          float* __restrict__ out)
{
    __shared__ float smn[512];
    __shared__ float smx[512];
    __shared__ float s_lo, s_hi;

    const int tid = threadIdx.x;
    const int TOT = BATCH * NBIN;                     // 4112

    float vals[9];
    int   cnt = 0;
    float mn = INFINITY, mx = -INFINITY;

    for (int i = tid; i < TOT; i += 512) {
        const int b = i / NBIN;
        const int l = i - b * NBIN;
        const float* p = partial + (size_t)b * BLOCKS_PER_BATCH * NBIN + l;
        float s = 0.0f;
#pragma unroll
        for (int k = 0; k < BLOCKS_PER_BATCH; ++k) s += p[k * NBIN];
        const float prof = s / mask_n[l];
        vals[cnt++] = prof;
        mn = fminf(mn, prof);
        mx = fmaxf(mx, prof);
    }

    smn[tid] = mn;
    smx[tid] = mx;
    __syncthreads();
    for (int s = 256; s > 0; s >>= 1) {
        if (tid < s) {
            smn[tid] = fminf(smn[tid], smn[tid + s]);
            smx[tid] = fmaxf(smx[tid], smx[tid + s]);
        }
        __syncthreads();
    }
    if (tid == 0) { s_lo = smn[0]; s_hi = smx[0]; }
    __syncthreads();

    const float lo = s_lo;
    const float range = s_hi - lo;
    cnt = 0;
    for (int i = tid; i < TOT; i += 512)
        out[i] = (vals[cnt++] - lo) / range;
}

// ---------------------------------------------------------------------------
extern "C" void kernel_launch(void* const* d_in, const int* in_sizes, int n_in,
                              void* d_out, int out_size, void* d_ws, size_t ws_size,
                              hipStream_t stream)
{
    const float* x      = (const float*)d_in[0];   // [16,3,512,512] f32
    // d_in[1] (dense one-hot mask, 269 MB) intentionally unused: recomputed
    // analytically on the fly, saving ~6x HBM traffic.
    const float* mask_n = (const float*)d_in[2];   // [257] f32

    float* partial = (float*)d_ws;                 // 16*32*257 f32 = 526 KB
    float* out     = (float*)d_out;                // [16,257] f32

    rect_profile_kernel<<<BATCH * BLOCKS_PER_BATCH, 256, 0, stream>>>(x, partial);
    rect_finalize_kernel<<<1, 512, 0, stream>>>(partial, mask_n, out);
}